// VQC_16063177687422
// MI455X (gfx1250) — compile-verified
//
#include <hip/hip_runtime.h>
#include <math.h>

typedef float v2f __attribute__((ext_vector_type(2)));
typedef float v8f __attribute__((ext_vector_type(8)));

#define DIM 16   // 2^4 amplitudes

// ---------------------------------------------------------------------------
// Kernel 1: build the fixed 16x16 complex unitary M from theta.
// The whole variational part (2 layers of CNOT ring + Rz,Ry,Rz) is batch-
// independent, so we materialize it once. One wave; lane t computes column t
// (circuit applied to basis state e_t), amplitudes held in registers.
// Bit convention: qubit q <-> bit (3-q) of the flattened index (reshape order).
// ---------------------------------------------------------------------------
__global__ void __launch_bounds__(32)
vqc_build_unitary(const float* __restrict__ theta, float* __restrict__ Mmat)
{
    const int t = threadIdx.x;
    if (t >= DIM) return;

    float re[DIM], im[DIM];
#pragma unroll
    for (int i = 0; i < DIM; ++i) { re[i] = (i == t) ? 1.f : 0.f; im[i] = 0.f; }

#pragma unroll
    for (int l = 0; l < 2; ++l) {
        // CNOT ring: (c,t) = (0,1),(1,2),(2,3),(3,0). Swap t=0/t=1 amps where c=1.
#pragma unroll
        for (int q = 0; q < 4; ++q) {
            const int cm = 8 >> q;
            const int tm = 8 >> ((q + 1) & 3);
#pragma unroll
            for (int i = 0; i < DIM; ++i) {
                if ((i & cm) && !(i & tm)) {
                    const int ip = i | tm;
                    float tr = re[i]; re[i] = re[ip]; re[ip] = tr;
                    float ti = im[i]; im[i] = im[ip]; im[ip] = ti;
                }
            }
        }
        // Per qubit: Rz(th0) then Ry(th1) then Rz(th2)
#pragma unroll
        for (int q = 0; q < 4; ++q) {
            const int m = 8 >> q;
            const float th0 = theta[(l * 4 + q) * 3 + 0];
            const float th1 = theta[(l * 4 + q) * 3 + 1];
            const float th2 = theta[(l * 4 + q) * 3 + 2];
            // Rz(th0): amp *= exp(i*phi), phi = (bit? +th/2 : -th/2)
            {
                const float c = cosf(th0 * 0.5f), s = sinf(th0 * 0.5f);
#pragma unroll
                for (int i = 0; i < DIM; ++i) {
                    const float sp = (i & m) ? s : -s;
                    const float r  = re[i] * c - im[i] * sp;
                    const float ii = im[i] * c + re[i] * sp;
                    re[i] = r; im[i] = ii;
                }
            }
            // Ry(th1): real rotation [[c,-s],[s,c]] on (bit=0, bit=1) pairs
            {
                const float c = cosf(th1 * 0.5f), s = sinf(th1 * 0.5f);
#pragma unroll
                for (int i = 0; i < DIM; ++i) {
                    if (!(i & m)) {
                        const int ip = i | m;
                        const float r0 = c * re[i] - s * re[ip];
                        const float r1 = s * re[i] + c * re[ip];
                        const float i0 = c * im[i] - s * im[ip];
                        const float i1 = s * im[i] + c * im[ip];
                        re[i] = r0; re[ip] = r1; im[i] = i0; im[ip] = i1;
                    }
                }
            }
            // Rz(th2)
            {
                const float c = cosf(th2 * 0.5f), s = sinf(th2 * 0.5f);
#pragma unroll
                for (int i = 0; i < DIM; ++i) {
                    const float sp = (i & m) ? s : -s;
                    const float r  = re[i] * c - im[i] * sp;
                    const float ii = im[i] * c + re[i] * sp;
                    re[i] = r; im[i] = ii;
                }
            }
        }
    }

    // M[row r][col t] = amplitude r of U|e_t>
    float2* M2 = (float2*)Mmat;
#pragma unroll
    for (int r = 0; r < DIM; ++r)
        M2[r * DIM + t] = make_float2(re[r], im[r]);
}

// ---------------------------------------------------------------------------
// Kernel 2: one wave handles 16 batch elements per tile via
// V_WMMA_F32_16X16X4_F32:  D[amp m, batch n] = sum_j M[m,j] * in[j,n]
// Complex split (f32 WMMA has no A/B NEG, so carry -M_im explicitly):
//   out_re = M_re*in_re + (-M_im)*in_im
//   out_im = M_im*in_re +   M_re *in_im
// Fragment layouts (wave32, ISA 7.12.2): half = lane>>4, n = lane&15
//   A 16x4 chunk kk:  row = n,  K = kk*4 + vgpr + 2*half
//   B 4x16 chunk kk:  col = n,  K = kk*4 + vgpr + 2*half
//   C/D 16x16:        col = n,  row m = vgpr + 8*half
// ---------------------------------------------------------------------------
__global__ void __launch_bounds__(256)
vqc_main(const float* __restrict__ x, const float* __restrict__ Mmat,
         float* __restrict__ out, int batch)
{
    const int lane  = threadIdx.x & 31;
    const int half  = lane >> 4;
    const int n     = lane & 15;
    const int wave  = (blockIdx.x * blockDim.x + threadIdx.x) >> 5;
    const int nwave = (gridDim.x * blockDim.x) >> 5;

    // Loop-invariant A fragments for M_re, M_im, -M_im.
    v2f Are[4], Aim[4], Ain[4];
    const float2* M2 = (const float2*)Mmat;
#pragma unroll
    for (int kk = 0; kk < 4; ++kk) {
#pragma unroll
        for (int v = 0; v < 2; ++v) {
            const int j = kk * 4 + v + half * 2;
            const float2 mv = M2[n * DIM + j];
            Are[kk][v] =  mv.x;
            Aim[kk][v] =  mv.y;
            Ain[kk][v] = -mv.y;
        }
    }

    const int ntiles = batch >> 4;
    for (int tile = wave; tile < ntiles; tile += nwave) {
        const int b = tile * 16 + n;

        // RX-encoded product state: amp(j) = (-i)^popc(j) * prod_q (bit? s_q : c_q)
        const float4 xv = ((const float4*)x)[b];
        float c[4], s[4];
        c[0] = __cosf(xv.x * 0.5f); s[0] = __sinf(xv.x * 0.5f);
        c[1] = __cosf(xv.y * 0.5f); s[1] = __sinf(xv.y * 0.5f);
        c[2] = __cosf(xv.z * 0.5f); s[2] = __sinf(xv.z * 0.5f);
        c[3] = __cosf(xv.w * 0.5f); s[3] = __sinf(xv.w * 0.5f);

        v2f Bre[4], Bim[4];
#pragma unroll
        for (int kk = 0; kk < 4; ++kk) {
#pragma unroll
            for (int v = 0; v < 2; ++v) {
                const int j = kk * 4 + v + half * 2;
                const float mag = ((j & 8) ? s[0] : c[0]) * ((j & 4) ? s[1] : c[1]) *
                                  ((j & 2) ? s[2] : c[2]) * ((j & 1) ? s[3] : c[3]);
                const int pc = __popc(j) & 3;
                Bre[kk][v] = (pc == 0) ? mag : ((pc == 2) ? -mag : 0.f);
                Bim[kk][v] = (pc == 3) ? mag : ((pc == 1) ? -mag : 0.f);
            }
        }

        v8f accRe = {};
        v8f accIm = {};
#pragma unroll
        for (int kk = 0; kk < 4; ++kk) {
            accRe = __builtin_amdgcn_wmma_f32_16x16x4_f32(
                false, Are[kk], false, Bre[kk], (short)0, accRe, false, false);
            accRe = __builtin_amdgcn_wmma_f32_16x16x4_f32(
                false, Ain[kk], false, Bim[kk], (short)0, accRe, false, false);
            accIm = __builtin_amdgcn_wmma_f32_16x16x4_f32(
                false, Aim[kk], false, Bre[kk], (short)0, accIm, false, false);
            accIm = __builtin_amdgcn_wmma_f32_16x16x4_f32(
                false, Are[kk], false, Bim[kk], (short)0, accIm, false, false);
        }

        // probs -> <Z_q>. Lane holds m = d + 8*half for d=0..7.
        // qubit0 <-> bit3 (= half), qubit1 <-> d&4, qubit2 <-> d&2, qubit3 <-> d&1
        float e0 = 0.f, e1 = 0.f, e2 = 0.f, e3 = 0.f;
#pragma unroll
        for (int d = 0; d < 8; ++d) {
            const float p = accRe[d] * accRe[d] + accIm[d] * accIm[d];
            e0 += p;
            e1 += (d & 4) ? -p : p;
            e2 += (d & 2) ? -p : p;
            e3 += (d & 1) ? -p : p;
        }
        if (half) e0 = -e0;
        e0 += __shfl_xor(e0, 16, 32);
        e1 += __shfl_xor(e1, 16, 32);
        e2 += __shfl_xor(e2, 16, 32);
        e3 += __shfl_xor(e3, 16, 32);

        if (half == 0)
            ((float4*)out)[b] = make_float4(e0, e1, e2, e3);
    }
}

extern "C" void kernel_launch(void* const* d_in, const int* in_sizes, int n_in,
                              void* d_out, int out_size, void* d_ws, size_t ws_size,
                              hipStream_t stream) {
    const float* x     = (const float*)d_in[0];   // (BATCH, 4) float32
    const float* theta = (const float*)d_in[1];   // (2, 4, 3) float32
    float* out  = (float*)d_out;                  // (BATCH, 4) float32
    float* Mmat = (float*)d_ws;                   // 16*16 float2 = 2 KB

    const int batch = in_sizes[0] / 4;

    vqc_build_unitary<<<1, 32, 0, stream>>>(theta, Mmat);

    const int threads = 256;                      // 8 waves / block
    const int blocks  = 2048;                     // 16384 waves, ~2 tiles each
    vqc_main<<<blocks, threads, 0, stream>>>(x, Mmat, out, batch);
}